// Seq2Seq_69844758168211
// MI455X (gfx1250) — compile-verified
//
#include <hip/hip_runtime.h>
#include <math.h>

// ---------------------------------------------------------------------------
// Seq2Seq GRU (enc 64-layer + dec 64-layer + MLP head) for MI455X / gfx1250.
//
// Latency-bound wavefront -> one persistent workgroup per GRU layer
// (64 blocks, 8 wave32s). Layers pipeline timesteps via a depth-8 ring buffer
// in global memory guarded by agent-scope acquire/release progress counters.
// Per-layer weights live in LDS (~298KB of the WGP's 320KB) for all 256 steps.
// Matmuls use exact-fp32 V_WMMA_F32_16X16X4_F32. A-fragments are cached in
// registers across each wave's 3 N-tiles, and every LDS matrix leading dim is
// padded +4 floats so fragment loads spread across all 64 LDS banks.
// ---------------------------------------------------------------------------

#define B_    64
#define T_    256
#define F_    128
#define H_    64
#define L_    64
#define G3    192          // 3*H
#define DEPTH 8            // inter-layer ring buffer depth (power of 2)

// Padded LDS leading dimensions (+4 floats -> conflict-free fragment loads).
#define LDF   132          // F_ + 4
#define LDH   68           // H_ + 4
#define LDG   196          // G3 + 4

typedef float v2f __attribute__((ext_vector_type(2)));
typedef float v8f __attribute__((ext_vector_type(8)));

__device__ __forceinline__ v8f wmma_f32_k4(v2f a, v2f b, v8f c) {
  // D = A(16x4) * B(4x16) + C, full fp32 (matches fp32 reference numerics)
  return __builtin_amdgcn_wmma_f32_16x16x4_f32(false, a, false, b, (short)0, c,
                                               false, false);
}

// Store C/D frag (16x16 f32): VGPR i -> row i (lanes 0-15) / row i+8 (16-31).
__device__ __forceinline__ void store_c(float* D, int ldd, v8f c, int mt,
                                        int nt, int lane) {
  const int col = nt * 16 + (lane & 15);
  const int r0  = mt * 16 + ((lane >> 4) << 3);
#pragma unroll
  for (int i = 0; i < 8; ++i) D[(r0 + i) * ldd + col] = c[i];
}

// G[mt, nt0..nt0+2] = A(64xK, ld=K+4) * W(NxK, ld=K+4)^T.
// A-fragments for the whole K-loop are cached in registers and reused
// across the wave's 3 N-tiles (halves LDS issue in the WMMA hot loop).
template <int K>
__device__ __forceinline__ void gemm_row3(const float* A, const float* W,
                                          int mt, int nt0, float* G, int lane) {
  constexpr int LDA = K + 4;
  constexpr int NK  = K / 4;
  v2f afr[NK];
  const float* pa = A + (mt * 16 + (lane & 15)) * LDA + ((lane >> 4) << 1);
#pragma unroll
  for (int k = 0; k < NK; ++k) {
    afr[k][0] = pa[4 * k];
    afr[k][1] = pa[4 * k + 1];
  }
#pragma unroll
  for (int ni = 0; ni < 3; ++ni) {
    const int nt = nt0 + ni;
    const float* pb = W + (nt * 16 + (lane & 15)) * LDA + ((lane >> 4) << 1);
    v8f acc = {};
#pragma unroll
    for (int k = 0; k < NK; ++k) {
      v2f b;
      b[0] = pb[4 * k];
      b[1] = pb[4 * k + 1];
      acc = wmma_f32_k4(afr[k], b, acc);
    }
    store_c(G, LDG, acc, mt, nt, lane);
  }
}

// Single-tile GEMM (MLP): D(mt,nt) = A(64xK) * W(NxK)^T, padded strides.
template <int K>
__device__ __forceinline__ v8f gemm_tile(const float* A, const float* W,
                                         int mt, int nt, int lane) {
  constexpr int LDA = K + 4;
  v8f acc = {};
  const float* pa = A + (mt * 16 + (lane & 15)) * LDA + ((lane >> 4) << 1);
  const float* pb = W + (nt * 16 + (lane & 15)) * LDA + ((lane >> 4) << 1);
#pragma unroll
  for (int k = 0; k < K / 4; ++k) {
    v2f a, b;
    a[0] = pa[4 * k]; a[1] = pa[4 * k + 1];
    b[0] = pb[4 * k]; b[1] = pb[4 * k + 1];
    acc = wmma_f32_k4(a, b, acc);
  }
  return acc;
}

// Stage packed rows x IN global matrix into LDS with stride IN+4.
template <int IN>
__device__ __forceinline__ void stage_mat(float* dst, const float* src,
                                          int rows, int tid, int nthr) {
  for (int i = tid; i < rows * IN / 4; i += nthr) {
    const int idx = i << 2;
    const int r = idx / IN, c = idx % IN;  // IN is a power of two
    *(float4*)(dst + r * (IN + 4) + c) = *(const float4*)(src + idx);
  }
}

__global__ void init_flags_kernel(int* prog) {
  const int i = threadIdx.x;
  if (i < 2 * L_) prog[i] = 0;
}

// One block per layer; blocks pipeline timesteps through `ybuf`/`prog`.
template <bool DEC>
__global__ void gru_pass_kernel(const float* __restrict__ context,
                                const float* __restrict__ Wi0,
                                const float* __restrict__ Wh0,
                                const float* __restrict__ bi0,
                                const float* __restrict__ bh0,
                                const float* __restrict__ Wis,
                                const float* __restrict__ Whs,
                                const float* __restrict__ bis,
                                const float* __restrict__ bhs,
                                const float* __restrict__ h_init, // dec: h_n
                                float* __restrict__ ybuf,  // [L][DEPTH][B][H]
                                int* __restrict__ prog,    // [L]
                                float* __restrict__ top_out) {
  extern __shared__ float smem[];
  const int l    = blockIdx.x;
  const int tid  = threadIdx.x;
  const int lane = tid & 31;
  const int wave = tid >> 5;

  float* sWi = smem;               // 192 x LDF (max)
  float* sWh = sWi + G3 * LDF;     // 192 x LDH
  float* sX  = sWh + G3 * LDH;     // 64 x LDF (max)
  float* sGi = sX + B_ * LDF;      // 64 x LDG
  float* sGh = sGi + B_ * LDG;     // 64 x LDG
  float* sH  = sGh + B_ * LDG;     // 64 x LDH

  const float* Wi = (l == 0) ? Wi0 : (Wis + (size_t)(l - 1) * G3 * H_);
  const float* Wh = (l == 0) ? Wh0 : (Whs + (size_t)(l - 1) * G3 * H_);
  const float* bi = (l == 0) ? bi0 : (bis + (size_t)(l - 1) * G3);
  const float* bh = (l == 0) ? bh0 : (bhs + (size_t)(l - 1) * G3);

  // Stage this layer's weights into LDS once (reused for all T_ steps).
  if (l == 0) stage_mat<F_>(sWi, Wi, G3, tid, blockDim.x);
  else        stage_mat<H_>(sWi, Wi, G3, tid, blockDim.x);
  stage_mat<H_>(sWh, Wh, G3, tid, blockDim.x);

  // Each thread owns 16 elements of the hidden state: row eb, cols ej..ej+15.
  const int eb = tid >> 2;
  const int ej = (tid & 3) << 4;
#pragma unroll
  for (int e = 0; e < 16; ++e)
    sH[eb * LDH + ej + e] = DEC ? h_init[eb * H_ + ej + e] : 0.0f;

  float accv[16];   // encoder top layer: running time-sum for mean pool
#pragma unroll
  for (int e = 0; e < 16; ++e) accv[e] = 0.0f;

  __syncthreads();

  for (int t = 0; t < T_; ++t) {
    // ---- pipeline flow control (thread 0 spins, agent-scope acquire) ----
    if (tid == 0) {
      if (l > 0) {
        while (__hip_atomic_load(&prog[l - 1], __ATOMIC_ACQUIRE,
                                 __HIP_MEMORY_SCOPE_AGENT) < t + 1)
          __builtin_amdgcn_s_sleep(1);
      }
      if (l < L_ - 1 && t >= DEPTH) {  // ring-buffer back-pressure
        while (__hip_atomic_load(&prog[l + 1], __ATOMIC_ACQUIRE,
                                 __HIP_MEMORY_SCOPE_AGENT) < t - DEPTH + 1)
          __builtin_amdgcn_s_sleep(1);
      }
    }
    __syncthreads();

    // ---- stage x_t (B x in) into LDS (padded stride) ----
    if (l == 0) {
      if (DEC && t == 0) {
        for (int i = tid; i < (B_ * F_) / 4; i += blockDim.x) {
          const int idx = i << 2;
          const int b = idx >> 7, k = idx & (F_ - 1);
          *(float4*)(sX + b * LDF + k) = make_float4(-1.f, -1.f, -1.f, -1.f);
        }
      } else {
        const int ts = DEC ? (t - 1) : t;  // decoder input = shifted context
        for (int i = tid; i < (B_ * F_) / 4; i += blockDim.x) {
          const int idx = i << 2;
          const int b = idx >> 7, k = idx & (F_ - 1);
          *(float4*)(sX + b * LDF + k) =
              *(const float4*)(context + ((size_t)b * T_ + ts) * F_ + k);
        }
      }
    } else {
      const float* ysrc = ybuf + ((size_t)(l - 1) * DEPTH +
                                  (t & (DEPTH - 1))) * B_ * H_;
      for (int i = tid; i < (B_ * H_) / 4; i += blockDim.x) {
        const int idx = i << 2;
        const int b = idx >> 6, k = idx & (H_ - 1);
        *(float4*)(sX + b * LDH + k) = *(const float4*)(ysrc + idx);
      }
    }
    __syncthreads();

    // ---- gi = x Wi^T, gh = h Wh^T via fp32 WMMA; 6 tiles per wave ----
    {
      const int wm = wave >> 2;  // m-tiles {2wm, 2wm+1}
      const int nt0 = (wave & 3) * 3;  // n-tiles {nt0 .. nt0+2}
#pragma unroll
      for (int mi = 0; mi < 2; ++mi) {
        const int mt = wm * 2 + mi;
        if (l == 0) gemm_row3<F_>(sX, sWi, mt, nt0, sGi, lane);
        else        gemm_row3<H_>(sX, sWi, mt, nt0, sGi, lane);
        gemm_row3<H_>(sH, sWh, mt, nt0, sGh, lane);
      }
    }
    __syncthreads();

    // ---- fused GRU gate math (PyTorch gate order r,z,n) ----
    float* ydst = (l < L_ - 1)
                      ? (ybuf + ((size_t)l * DEPTH + (t & (DEPTH - 1))) * B_ * H_)
                      : nullptr;
#pragma unroll
    for (int e = 0; e < 16; ++e) {
      const int j = ej + e;
      const float hprev = sH[eb * LDH + j];
      const float ir  = sGi[eb * LDG + j]          + bi[j];
      const float iz  = sGi[eb * LDG + H_ + j]     + bi[H_ + j];
      const float in_ = sGi[eb * LDG + 2 * H_ + j] + bi[2 * H_ + j];
      const float hr  = sGh[eb * LDG + j]          + bh[j];
      const float hz  = sGh[eb * LDG + H_ + j]     + bh[H_ + j];
      const float hn  = sGh[eb * LDG + 2 * H_ + j] + bh[2 * H_ + j];
      const float r = 1.0f / (1.0f + __expf(-(ir + hr)));
      const float z = 1.0f / (1.0f + __expf(-(iz + hz)));
      const float n = tanhf(in_ + r * hn);
      const float hnew = (1.0f - z) * n + z * hprev;
      sH[eb * LDH + j] = hnew;
      if (l < L_ - 1)      ydst[eb * H_ + j] = hnew;
      else if (DEC)        top_out[((size_t)t * B_ + eb) * H_ + j] = hnew;
      else                 accv[e] += hnew;   // encoder mean pool
    }
    __threadfence();  // y_t visible device-wide before publishing progress
    __syncthreads();
    if (tid == 0)
      __hip_atomic_store(&prog[l], t + 1, __ATOMIC_RELEASE,
                         __HIP_MEMORY_SCOPE_AGENT);
  }

  if (!DEC && l == L_ - 1) {  // h_n = mean over time of top-layer outputs
    const float inv = 1.0f / (float)T_;
#pragma unroll
    for (int e = 0; e < 16; ++e)
      top_out[eb * H_ + ej + e] = accv[e] * inv;
  }
}

// out = relu(dec_top @ W1^T + b1) @ W2^T + b2, scattered to (B,T,F).
__global__ void mlp_kernel(const float* __restrict__ dec_top,  // [T*B, H]
                           const float* __restrict__ W1, const float* __restrict__ b1,
                           const float* __restrict__ W2, const float* __restrict__ b2,
                           float* __restrict__ out) {            // [B, T, F]
  extern __shared__ float smem[];
  float* sW1 = smem;               // 64 x LDH
  float* sW2 = sW1 + H_ * LDH;     // 128 x LDH
  float* sX  = sW2 + F_ * LDH;     // 64 x LDH
  float* sH1 = sX + 64 * LDH;      // 64 x LDH
  const int tid = threadIdx.x, lane = tid & 31, wave = tid >> 5;
  const int r0 = blockIdx.x * 64;

  stage_mat<H_>(sW1, W1, H_, tid, blockDim.x);
  stage_mat<H_>(sW2, W2, F_, tid, blockDim.x);
  stage_mat<H_>(sX, dec_top + (size_t)r0 * H_, 64, tid, blockDim.x);
  __syncthreads();

  // h1 = relu(x W1^T + b1): 4x4 tile grid, 2 tiles per wave.
#pragma unroll
  for (int i = 0; i < 2; ++i) {
    const int tile = wave * 2 + i;
    const int mt = tile >> 2, nt = tile & 3;
    v8f c = gemm_tile<H_>(sX, sW1, mt, nt, lane);
    const int col = nt * 16 + (lane & 15);
    const int rb  = mt * 16 + ((lane >> 4) << 3);
    const float bv = b1[col];
#pragma unroll
    for (int q = 0; q < 8; ++q) {
      const float v = c[q] + bv;
      sH1[(rb + q) * LDH + col] = v > 0.f ? v : 0.f;
    }
  }
  __syncthreads();

  // out = h1 W2^T + b2: 4x8 tile grid, 4 tiles per wave; remap row t*B+b.
#pragma unroll
  for (int i = 0; i < 4; ++i) {
    const int tile = wave * 4 + i;
    const int mt = tile >> 3, nt = tile & 7;
    v8f c = gemm_tile<H_>(sH1, sW2, mt, nt, lane);
    const int col = nt * 16 + (lane & 15);
    const int rb  = mt * 16 + ((lane >> 4) << 3);
    const float bv = b2[col];
#pragma unroll
    for (int q = 0; q < 8; ++q) {
      const int grow = r0 + rb + q;           // = t*B + b
      const int t = grow >> 6, bb = grow & 63;
      out[((size_t)bb * T_ + t) * F_ + col] = c[q] + bv;
    }
  }
}

extern "C" void kernel_launch(void* const* d_in, const int* in_sizes, int n_in,
                              void* d_out, int out_size, void* d_ws,
                              size_t ws_size, hipStream_t stream) {
  (void)in_sizes; (void)n_in; (void)out_size; (void)ws_size;
  const float* context = (const float*)d_in[0];
  const float* eWi0 = (const float*)d_in[1];
  const float* eWh0 = (const float*)d_in[2];
  const float* ebi0 = (const float*)d_in[3];
  const float* ebh0 = (const float*)d_in[4];
  const float* eWis = (const float*)d_in[5];
  const float* eWhs = (const float*)d_in[6];
  const float* ebis = (const float*)d_in[7];
  const float* ebhs = (const float*)d_in[8];
  const float* dWi0 = (const float*)d_in[9];
  const float* dWh0 = (const float*)d_in[10];
  const float* dbi0 = (const float*)d_in[11];
  const float* dbh0 = (const float*)d_in[12];
  const float* dWis = (const float*)d_in[13];
  const float* dWhs = (const float*)d_in[14];
  const float* dbis = (const float*)d_in[15];
  const float* dbhs = (const float*)d_in[16];
  const float* W1 = (const float*)d_in[17];
  const float* b1 = (const float*)d_in[18];
  const float* W2 = (const float*)d_in[19];
  const float* b2 = (const float*)d_in[20];

  // Workspace carve-up (~12.6 MB).
  char* ws = (char*)d_ws;
  float* ybuf = (float*)ws;
  size_t o = (size_t)L_ * DEPTH * B_ * H_ * sizeof(float);
  float* dec_top = (float*)(ws + o); o += (size_t)T_ * B_ * H_ * sizeof(float);
  float* h_n     = (float*)(ws + o); o += (size_t)B_ * H_ * sizeof(float);
  int*   prog    = (int*)(ws + o);

  constexpr int SH_GRU =
      (G3 * LDF + G3 * LDH + B_ * LDF + 2 * B_ * LDG + B_ * LDH) *
      (int)sizeof(float);                                        // 305,152 B
  constexpr int SH_MLP = (4 * 64 * LDH + F_ * LDH) * (int)sizeof(float) -
                         (64 * LDH) * (int)sizeof(float);        // 87,040 B

  // Opt in to >64KB dynamic LDS (CDNA5 WGP has 320KB). Idempotent.
  hipFuncSetAttribute(reinterpret_cast<const void*>(gru_pass_kernel<false>),
                      hipFuncAttributeMaxDynamicSharedMemorySize, SH_GRU);
  hipFuncSetAttribute(reinterpret_cast<const void*>(gru_pass_kernel<true>),
                      hipFuncAttributeMaxDynamicSharedMemorySize, SH_GRU);
  hipFuncSetAttribute(reinterpret_cast<const void*>(mlp_kernel),
                      hipFuncAttributeMaxDynamicSharedMemorySize, SH_MLP);

  init_flags_kernel<<<1, 128, 0, stream>>>(prog);

  // Encoder wavefront: one block per layer; top layer emits h_n (mean pool).
  gru_pass_kernel<false><<<dim3(L_), dim3(256), SH_GRU, stream>>>(
      context, eWi0, eWh0, ebi0, ebh0, eWis, eWhs, ebis, ebhs,
      nullptr, ybuf, prog, h_n);

  // Decoder wavefront: h0 = h_n for every layer; top layer emits dec_top.
  gru_pass_kernel<true><<<dim3(L_), dim3(256), SH_GRU, stream>>>(
      context, dWi0, dWh0, dbi0, dbh0, dWis, dWhs, dbis, dbhs,
      h_n, ybuf, prog + L_, dec_top);

  // Output MLP over all T*B rows.
  mlp_kernel<<<dim3((T_ * B_) / 64), dim3(256), SH_MLP, stream>>>(
      dec_top, W1, b1, W2, b2, (float*)d_out);
}